// SMPL_76579266888264
// MI455X (gfx1250) — compile-verified
//
#include <hip/hip_runtime.h>
#include <math.h>

#define NB   4096      // batch
#define NV   778       // vertices
#define NJ   16        // joints
#define C3   2334      // V*3
#define KTOT 145       // 10 beta + 135 pose
#define KP   148       // K padded to multiple of 4
#define CP   2368      // cols padded to multiple of 64

typedef __attribute__((ext_vector_type(2))) float v2f;
typedef __attribute__((ext_vector_type(8))) float v8f;
typedef int async_int4 __attribute__((vector_size(16)));   // matches builtin param

#if defined(__gfx1250__) && __has_builtin(__builtin_amdgcn_wmma_f32_16x16x4_f32)
#define HAVE_WMMA_F32X4 1
#else
#define HAVE_WMMA_F32X4 0
#endif

#if defined(__gfx1250__) && __has_builtin(__builtin_amdgcn_global_load_async_to_lds_b128) && \
    __has_builtin(__builtin_amdgcn_s_wait_asynccnt)
#define HAVE_ASYNC_LDS 1
#else
#define HAVE_ASYNC_LDS 0
#endif

// ---------------------------------------------------------------------------
// Pack [shapedirs(10 x 2334); posedirs(135 x 2334)] into a zero-padded,
// K-pair-interleaved [KP/2][CP][2] layout: element (k,col) lives at
// ((k>>1)*CP + col)*2 + (k&1). This makes each WMMA B-fragment
// {SD[k][col], SD[k+1][col]} a single aligned 8-byte load. 1.4 MB -> L2.
// ---------------------------------------------------------------------------
__global__ __launch_bounds__(256) void k_pack_sd(
    const float* __restrict__ shapedirs, const float* __restrict__ posedirs,
    float* __restrict__ sd)
{
  int idx = blockIdx.x * 256 + threadIdx.x;
  if (idx >= KP * CP) return;
  int k = idx / CP, col = idx % CP;
  float v = 0.f;
  if (col < C3 && k < KTOT)
    v = (k < 10) ? shapedirs[(size_t)k * C3 + col]
                 : posedirs[(size_t)(k - 10) * C3 + col];
  sd[((size_t)(k >> 1) * CP + col) * 2 + (k & 1)] = v;
}

// ---------------------------------------------------------------------------
// Factor joint regression through shape space:
//   J(n) = J0 + JS * beta(n),   JS[(j*3+c)*10+b] = sum_v Jreg[v,j]*shapedirs[b, v*3+c]
// 528 dot products of length 778; one block.
// ---------------------------------------------------------------------------
__global__ __launch_bounds__(512) void k_precompute_js(
    const float* __restrict__ Jreg, const float* __restrict__ shapedirs,
    const float* __restrict__ v_template,
    float* __restrict__ js, float* __restrict__ j0)
{
  for (int out = threadIdx.x; out < 528; out += blockDim.x) {
    float s = 0.f;
    if (out < 480) {
      int b = out % 10, rem = out / 10;
      int c = rem % 3, j = rem / 3;
      for (int v = 0; v < NV; ++v)
        s += Jreg[v * NJ + j] * shapedirs[(size_t)b * C3 + v * 3 + c];
      js[out] = s;
    } else {
      int q = out - 480, c = q % 3, j = q / 3;
      for (int v = 0; v < NV; ++v)
        s += Jreg[v * NJ + j] * v_template[v * 3 + c];
      j0[q] = s;
    }
  }
}

// ---------------------------------------------------------------------------
// Per-body: Rodrigues -> Rs (output 3) + GEMM coefficient row
//           [beta(10) | Rs[1:]-I (135) | 0 0 0], joint chain -> A matrices.
// One thread per body; chain kept fully in registers (parents are i-1 except
// chain heads {1,4,7,10,13} whose parent is the root).
// ---------------------------------------------------------------------------
__global__ __launch_bounds__(256) void k_pern(
    const float* __restrict__ beta, const float* __restrict__ theta,
    const float* __restrict__ js, const float* __restrict__ j0,
    float* __restrict__ wc, float* __restrict__ wA, float* __restrict__ Rs_out)
{
  int n = blockIdx.x * blockDim.x + threadIdx.x;
  if (n >= NB) return;

  float bet[10];
#pragma unroll
  for (int b = 0; b < 10; ++b) bet[b] = beta[(size_t)n * 10 + b];

  float* c = wc + (size_t)n * KP;
#pragma unroll
  for (int b = 0; b < 10; ++b) c[b] = bet[b];
  c[145] = 0.f; c[146] = 0.f; c[147] = 0.f;

  const float* th = theta + (size_t)n * 48;
  float* rs = Rs_out + (size_t)n * 144;

  float Groot[12], Gprev[12], Jroot[3], Jprev[3];
#pragma unroll
  for (int e = 0; e < 12; ++e) { Groot[e] = 0.f; Gprev[e] = 0.f; }
  Jroot[0]=Jroot[1]=Jroot[2]=0.f; Jprev[0]=Jprev[1]=Jprev[2]=0.f;

  for (int i = 0; i < 16; ++i) {
    // --- Rodrigues (reference-faithful: eps added inside the norm only) ---
    float tx = th[i*3+0], ty = th[i*3+1], tz = th[i*3+2];
    float sx = tx + 1e-8f, sy = ty + 1e-8f, sz = tz + 1e-8f;
    float ang = sqrtf(sx*sx + sy*sy + sz*sz);
    float inv = 1.f / ang;
    float rx = tx*inv, ry = ty*inv, rz = tz*inv;
    float h = 0.5f * ang;
    float sw = sinf(h), cw = cosf(h);
    float qw = cw, qx = sw*rx, qy = sw*ry, qz = sw*rz;
    float qn = rsqrtf(qw*qw + qx*qx + qy*qy + qz*qz);
    qw *= qn; qx *= qn; qy *= qn; qz *= qn;
    float w2=qw*qw, x2=qx*qx, y2=qy*qy, z2=qz*qz;
    float wx=qw*qx, wy=qw*qy, wz=qw*qz, xy=qx*qy, xz=qx*qz, yz=qy*qz;
    float R[9];
    R[0]=w2+x2-y2-z2; R[1]=2.f*(xy-wz);   R[2]=2.f*(xz+wy);
    R[3]=2.f*(xy+wz); R[4]=w2-x2+y2-z2;   R[5]=2.f*(yz-wx);
    R[6]=2.f*(xz-wy); R[7]=2.f*(yz+wx);   R[8]=w2-x2-y2+z2;
#pragma unroll
    for (int e = 0; e < 9; ++e) rs[i*9+e] = R[e];
    if (i >= 1) {
#pragma unroll
      for (int e = 0; e < 9; ++e)
        c[10 + (i-1)*9 + e] = R[e] - ((e==0||e==4||e==8) ? 1.f : 0.f);
    }

    // --- joint location: J_i = J0 + JS * beta ---
    float Ji[3];
#pragma unroll
    for (int cc = 0; cc < 3; ++cc) {
      float s = j0[i*3+cc];
#pragma unroll
      for (int b = 0; b < 10; ++b) s += js[(i*3+cc)*10 + b] * bet[b];
      Ji[cc] = s;
    }

    // --- rigid chain (rows 0..2 of the 4x4) ---
    float G[12];
    if (i == 0) {
#pragma unroll
      for (int r = 0; r < 3; ++r) {          // Rs0 * diag(1,-1,-1) | J0
        G[r*4+0] =  R[r*3+0];
        G[r*4+1] = -R[r*3+1];
        G[r*4+2] = -R[r*3+2];
        G[r*4+3] =  Ji[r];
      }
#pragma unroll
      for (int e = 0; e < 12; ++e) Groot[e] = G[e];
      Jroot[0]=Ji[0]; Jroot[1]=Ji[1]; Jroot[2]=Ji[2];
    } else {
      bool fromRoot = ((i - 1) % 3) == 0;    // i in {1,4,7,10,13}
      float Gp[12], Jp[3];
#pragma unroll
      for (int e = 0; e < 12; ++e) Gp[e] = fromRoot ? Groot[e] : Gprev[e];
#pragma unroll
      for (int e = 0; e < 3; ++e)  Jp[e] = fromRoot ? Jroot[e] : Jprev[e];
      float t0 = Ji[0]-Jp[0], t1 = Ji[1]-Jp[1], t2 = Ji[2]-Jp[2];
#pragma unroll
      for (int r = 0; r < 3; ++r) {
#pragma unroll
        for (int cc = 0; cc < 3; ++cc)
          G[r*4+cc] = Gp[r*4+0]*R[0*3+cc] + Gp[r*4+1]*R[1*3+cc] + Gp[r*4+2]*R[2*3+cc];
        G[r*4+3] = Gp[r*4+3] + Gp[r*4+0]*t0 + Gp[r*4+1]*t1 + Gp[r*4+2]*t2;
      }
#pragma unroll
      for (int e = 0; e < 12; ++e) Gprev[e] = G[e];
      Jprev[0]=Ji[0]; Jprev[1]=Ji[1]; Jprev[2]=Ji[2];
    }

    // --- A_i = G_i with translation column minus G_i * [J_i;0] ---
    float* Arow = wA + (size_t)n * 192 + i * 12;
#pragma unroll
    for (int r = 0; r < 3; ++r) {
      Arow[r*4+0] = G[r*4+0];
      Arow[r*4+1] = G[r*4+1];
      Arow[r*4+2] = G[r*4+2];
      Arow[r*4+3] = G[r*4+3] - (G[r*4+0]*Ji[0] + G[r*4+1]*Ji[1] + G[r*4+2]*Ji[2]);
    }
  }
}

// ---------------------------------------------------------------------------
// The real GEMM: D = coef (4096 x 148) * SD (148 x 2368) via
// V_WMMA_F32_16X16X4_F32; epilogue adds v_template -> v_posed.
// Block = 8 waves: 16 bodies x 512 cols; each wave owns a 16x64 tile (4 accs).
// Coefficient block is contiguous (rows nbase..nbase+15, stride KP), staged
// into LDS via GLOBAL_LOAD_ASYNC_TO_LDS_B128 when available (ASYNCcnt path).
// Fragment layouts per ISA 7.12.2:
//   A : lanes 0-15 hold K={0,1}, lanes 16-31 K={2,3}, row M = lane&15
//   B : N = lane&15, lanes 0-15 K rows {0,1}, lanes 16-31 {2,3}
//   D : VGPR p -> M = p + 8*(lane>=16), N = lane&15
// ---------------------------------------------------------------------------
__global__ __launch_bounds__(256) void k_gemm(
    const float* __restrict__ wc, const float* __restrict__ sd,
    const float* __restrict__ v_template, float* __restrict__ vposed)
{
  __shared__ __align__(16) float cl[16 * KP];
  int tid = threadIdx.x;
  int nbase = blockIdx.y << 4;

#if HAVE_ASYNC_LDS
  {
    // 16*KP floats = 592 16-byte packets, contiguous in global memory.
    float* gsrc = const_cast<float*>(wc) + (size_t)nbase * KP;
    for (int i = tid; i < (16 * KP) / 4; i += 256) {
      async_int4* g4 = (async_int4*)(gsrc + 4 * i);
      async_int4* l4 = (async_int4*)(cl + 4 * i);
      __builtin_amdgcn_global_load_async_to_lds_b128(
          (__attribute__((address_space(1))) async_int4*)g4,
          (__attribute__((address_space(3))) async_int4*)l4,
          0, 0);
    }
    __builtin_amdgcn_s_wait_asynccnt(0);
  }
#else
  {
    const float4* src4 = (const float4*)(wc + (size_t)nbase * KP);
    float4* dst4 = (float4*)cl;
    for (int i = tid; i < (16 * KP) / 4; i += 256) dst4[i] = src4[i];
  }
#endif
  __syncthreads();

  int lane = tid & 31, wave = tid >> 5;
  int colbase = blockIdx.x * 512 + wave * 64;
  if (colbase >= CP) return;

  int m16 = lane & 15;
  int hi  = lane >> 4;          // which half-wave
  int klo = hi << 1;            // K offset 0 or 2
  int mrow = hi << 3;           // D row offset 0 or 8
  const float* cm = cl + m16 * KP;
  int col = colbase + m16;
  const v2f* Bp = (const v2f*)sd;   // pair (k, k+1) per element

  v8f acc0 = {0,0,0,0,0,0,0,0};
  v8f acc1 = {0,0,0,0,0,0,0,0};
  v8f acc2 = {0,0,0,0,0,0,0,0};
  v8f acc3 = {0,0,0,0,0,0,0,0};

  for (int kb = 0; kb < KP; kb += 4) {
#if HAVE_WMMA_F32X4
    v2f a = { cm[kb + klo], cm[kb + klo + 1] };
    const v2f* Brow = Bp + (size_t)((kb + klo) >> 1) * CP + col;
    v2f b0 = Brow[0];
    v2f b1 = Brow[16];
    v2f b2 = Brow[32];
    v2f b3 = Brow[48];
    acc0 = __builtin_amdgcn_wmma_f32_16x16x4_f32(false, a, false, b0, (short)0, acc0, false, false);
    acc1 = __builtin_amdgcn_wmma_f32_16x16x4_f32(false, a, false, b1, (short)0, acc1, false, false);
    acc2 = __builtin_amdgcn_wmma_f32_16x16x4_f32(false, a, false, b2, (short)0, acc2, false, false);
    acc3 = __builtin_amdgcn_wmma_f32_16x16x4_f32(false, a, false, b3, (short)0, acc3, false, false);
#else
    // scalar fallback producing the same per-lane fragment layout
#pragma unroll
    for (int p = 0; p < 8; ++p) {
      const float* crow = cl + (p + mrow) * KP + kb;
#pragma unroll
      for (int kk = 0; kk < 4; ++kk) {
        int k = kb + kk;
        const float* Brow = sd + ((size_t)(k >> 1) * CP) * 2 + (k & 1);
        float av = crow[kk];
        acc0[p] += av * Brow[(col) * 2];
        acc1[p] += av * Brow[(col + 16) * 2];
        acc2[p] += av * Brow[(col + 32) * 2];
        acc3[p] += av * Brow[(col + 48) * 2];
      }
    }
#endif
  }

#pragma unroll
  for (int p = 0; p < 8; ++p) {
    size_t row = (size_t)(nbase + mrow + p) * CP;
    int c0 = col;
    float t0 = (c0      < C3) ? v_template[c0]      : 0.f;
    float t1 = (c0 + 16 < C3) ? v_template[c0 + 16] : 0.f;
    float t2 = (c0 + 32 < C3) ? v_template[c0 + 32] : 0.f;
    float t3 = (c0 + 48 < C3) ? v_template[c0 + 48] : 0.f;
    vposed[row + c0]      = acc0[p] + t0;
    vposed[row + c0 + 16] = acc1[p] + t1;
    vposed[row + c0 + 32] = acc2[p] + t2;
    vposed[row + c0 + 48] = acc3[p] + t3;
  }
}

// ---------------------------------------------------------------------------
// Per-vertex 16-way blend of A matrices + apply. A(n) staged in LDS.
// ---------------------------------------------------------------------------
__global__ __launch_bounds__(256) void k_blend(
    const float* __restrict__ wA, const float* __restrict__ vposed,
    const float* __restrict__ weights, float* __restrict__ verts)
{
  __shared__ float As[192];
  int n = blockIdx.y;
  if (threadIdx.x < 192) As[threadIdx.x] = wA[(size_t)n * 192 + threadIdx.x];
  __syncthreads();
  int v = blockIdx.x * 256 + threadIdx.x;
  if (v >= NV) return;

  float T[12];
#pragma unroll
  for (int e = 0; e < 12; ++e) T[e] = 0.f;
#pragma unroll
  for (int j = 0; j < 16; ++j) {
    float w = weights[(size_t)v * 16 + j];
#pragma unroll
    for (int e = 0; e < 12; ++e) T[e] += w * As[j * 12 + e];
  }
  const float* p = vposed + (size_t)n * CP + v * 3;
  float p0 = p[0], p1 = p[1], p2 = p[2];
  float* o = verts + (size_t)n * C3 + v * 3;
  o[0] = T[0]*p0 + T[1]*p1 + T[2]*p2  + T[3];
  o[1] = T[4]*p0 + T[5]*p1 + T[6]*p2  + T[7];
  o[2] = T[8]*p0 + T[9]*p1 + T[10]*p2 + T[11];
}

// ---------------------------------------------------------------------------
// joints = Jreg^T * verts per body: register partials + ds_add_f32 reduction.
// ---------------------------------------------------------------------------
__global__ __launch_bounds__(256) void k_joints(
    const float* __restrict__ verts, const float* __restrict__ Jreg,
    float* __restrict__ joints)
{
  __shared__ float acc[48];
  int n = blockIdx.x;
  if (threadIdx.x < 48) acc[threadIdx.x] = 0.f;
  __syncthreads();

  float loc[48];
#pragma unroll
  for (int e = 0; e < 48; ++e) loc[e] = 0.f;

  for (int v = threadIdx.x; v < NV; v += 256) {
    const float* vp = verts + (size_t)n * C3 + v * 3;
    float p0 = vp[0], p1 = vp[1], p2 = vp[2];
    const float* jr = Jreg + (size_t)v * 16;
#pragma unroll
    for (int j = 0; j < 16; ++j) {
      float w = jr[j];
      loc[j*3+0] += w * p0;
      loc[j*3+1] += w * p1;
      loc[j*3+2] += w * p2;
    }
  }
#pragma unroll
  for (int e = 0; e < 48; ++e) atomicAdd(&acc[e], loc[e]);
  __syncthreads();
  if (threadIdx.x < 48) joints[(size_t)n * 48 + threadIdx.x] = acc[threadIdx.x];
}

// ---------------------------------------------------------------------------
extern "C" void kernel_launch(void* const* d_in, const int* in_sizes, int n_in,
                              void* d_out, int out_size, void* d_ws, size_t ws_size,
                              hipStream_t stream) {
  (void)in_sizes; (void)n_in; (void)out_size; (void)ws_size;

  const float* beta       = (const float*)d_in[0];
  const float* theta      = (const float*)d_in[1];
  const float* v_template = (const float*)d_in[2];
  const float* shapedirs  = (const float*)d_in[3];
  const float* posedirs   = (const float*)d_in[4];
  const float* Jreg       = (const float*)d_in[5];
  const float* weights    = (const float*)d_in[6];

  float* out    = (float*)d_out;
  float* verts  = out;                              // N*V*3
  float* joints = verts + (size_t)NB * C3;          // N*16*3
  float* Rs     = joints + (size_t)NB * 48;         // N*16*9

  float* ws     = (float*)d_ws;
  float* wc     = ws;                               // N x KP coefficient rows
  float* wA     = wc + (size_t)NB * KP;             // N x 16 x 12 A matrices
  float* sd     = wA + (size_t)NB * 192;            // KP x CP packed dirs (paired)
  float* js     = sd + (size_t)KP * CP;             // 480
  float* j0     = js + 480;                         // 48
  float* vposed = j0 + 48;                          // N x CP

  k_pack_sd<<<(KP * CP + 255) / 256, 256, 0, stream>>>(shapedirs, posedirs, sd);
  k_precompute_js<<<1, 512, 0, stream>>>(Jreg, shapedirs, v_template, js, j0);
  k_pern<<<NB / 256, 256, 0, stream>>>(beta, theta, js, j0, wc, wA, Rs);
  k_gemm<<<dim3((CP + 511) / 512, NB / 16), 256, 0, stream>>>(wc, sd, v_template, vposed);
  k_blend<<<dim3((NV + 255) / 256, NB), 256, 0, stream>>>(wA, vposed, weights, verts);
  k_joints<<<NB, 256, 0, stream>>>(verts, Jreg, joints);
}